// final_extractor_59115929862513
// MI455X (gfx1250) — compile-verified
//
#include <hip/hip_runtime.h>
#include <cstdint>
#include <math.h>

// ---------------------------------------------------------------------------
// Masked max+mean pooling over S.  feats[64,2048,256] f32, mask[64,2048] f32.
// Roofline: 128 MiB stream @ 23.3 TB/s ~ 5.8us floor; ~67 MFLOP of fmax/fma
// is negligible -> pure HBM-bandwidth problem.  gfx1250-specific path used:
// GLOBAL_LOAD_ASYNC_TO_LDS_B128 deep prefetch (ASYNCcnt) into 320KB-WGP LDS,
// wave32 consume via ds_load_b128, s_wait_asynccnt pipelined drain.
// WMMA is intentionally unused: masked max has no matmul form and the mean
// as GEMM would waste 15/16 of the matrix pipe without moving the roofline.
// ---------------------------------------------------------------------------

#define PF 4            // prefetch depth (tiles in flight per wave)
#define SCHUNKS 16      // S split for occupancy: 64*2*16 = 2048 blocks

constexpr int kB = 64, kS = 2048, kD = 256;
constexpr int kSCLEN = kS / SCHUNKS;   // 128 rows per chunk
constexpr int kTILES = kSCLEN / 8;     // 16 tiles of 8 rows

__device__ __forceinline__ void async_load_b128(uint32_t lds_off, uint64_t gaddr) {
  // GV mode: vdst = LDS byte offset VGPR, vaddr = 64-bit global address.
  asm volatile("global_load_async_to_lds_b128 %0, %1, off"
               :: "v"(lds_off), "v"(gaddr)
               : "memory");
}

template <int N>
__device__ __forceinline__ void wait_asynccnt() {
  // inline asm (not builtin): memory clobber pins the subsequent ds_load of
  // the staged tile after the wait in both IR and MI scheduling.
  asm volatile("s_wait_asynccnt %0" :: "i"(N) : "memory");
}

__device__ __forceinline__ void wait_dscnt0() {
  // WAR guard: ASYNC and DS queues are mutually unordered; the LDS slot being
  // re-targeted must have finished its ds_load consumption first.
  asm volatile("s_wait_dscnt 0x0" ::: "memory");
}

// ---------------------------------------------------------------------------
// Pass 1: block = (batch b, d-chunk of 128, s-chunk of 128 rows).
// 256 threads = 8 row-groups x 32 cols; thread owns 4 d's of rows s0+row+8t.
// Async pipeline: PF tiles (8 rows x 128 d = 4KB) in flight per block.
// ---------------------------------------------------------------------------
__global__ __launch_bounds__(256)
void pool_pass1(const float* __restrict__ feats, const float* __restrict__ mask,
                float* __restrict__ psum, float* __restrict__ pmax) {
  __shared__ float4 stage[PF * 256];   // 16 KB async staging, per-thread slots
  __shared__ float  smask[kSCLEN];     // mask chunk (512 B)
  __shared__ float4 reds[256];
  __shared__ float4 redm[256];

  const int tid = threadIdx.x;
  const int col = tid & 31;            // 32 cols * float4 = 128 d
  const int row = tid >> 5;            // 8 interleaved s rows
  int bid = blockIdx.x;
  const int sc     = bid % SCHUNKS; bid /= SCHUNKS;
  const int dchunk = bid & 1;
  const int b      = bid >> 1;
  const int s0     = sc * kSCLEN;

  const float* fbase =
      feats + ((size_t)b * kS + s0 + row) * kD + dchunk * 128 + col * 4;

  // Stage mask chunk into LDS.
  if (tid < kSCLEN) smask[tid] = mask[(size_t)b * kS + s0 + tid];

  // LDS byte offset of this thread's slot 0 (low 32 bits of the generic
  // address of an LDS object == group-relative LDS offset).
  const uint32_t lds_slot0 =
      (uint32_t)(uintptr_t)(&stage[0]) + (uint32_t)tid * 16u;

  // Prime the pipeline: PF tiles in flight (ASYNCcnt == PF).
#pragma unroll
  for (int p = 0; p < PF; ++p)
    async_load_b128(lds_slot0 + (uint32_t)p * (256u * 16u),
                    (uint64_t)(uintptr_t)(fbase + (size_t)p * 8 * kD));

  __syncthreads();  // smask visible (does not drain ASYNCcnt)

  float4 vsum = make_float4(0.f, 0.f, 0.f, 0.f);
  float4 vmax = make_float4(-INFINITY, -INFINITY, -INFINITY, -INFINITY);

  auto consume = [&](int t) {
    float4 v = stage[(t & (PF - 1)) * 256 + tid];   // ds_load_b128
    float mk = smask[row + 8 * t];
    const bool on = mk > 0.f;
    vmax.x = fmaxf(vmax.x, on ? v.x : -INFINITY);
    vmax.y = fmaxf(vmax.y, on ? v.y : -INFINITY);
    vmax.z = fmaxf(vmax.z, on ? v.z : -INFINITY);
    vmax.w = fmaxf(vmax.w, on ? v.w : -INFINITY);
    vsum.x = fmaf(v.x, mk, vsum.x);
    vsum.y = fmaf(v.y, mk, vsum.y);
    vsum.z = fmaf(v.z, mk, vsum.z);
    vsum.w = fmaf(v.w, mk, vsum.w);
  };

  // Steady state: wait only for the oldest tile (async loads complete in
  // order per wave), consume it, reissue its slot PF tiles ahead.
  for (int t = 0; t < kTILES - PF; ++t) {
    wait_asynccnt<PF - 1>();
    consume(t);
    wait_dscnt0();
    async_load_b128(lds_slot0 + (uint32_t)(t & (PF - 1)) * (256u * 16u),
                    (uint64_t)(uintptr_t)(fbase + (size_t)(t + PF) * 8 * kD));
  }
  // Drain the last PF tiles with decreasing counts.
  wait_asynccnt<3>(); consume(kTILES - 4);
  wait_asynccnt<2>(); consume(kTILES - 3);
  wait_asynccnt<1>(); consume(kTILES - 2);
  wait_asynccnt<0>(); consume(kTILES - 1);

  // Reduce the 8 row-groups (threads sharing a col) via LDS tree.
  reds[tid] = vsum;
  redm[tid] = vmax;
  __syncthreads();
#pragma unroll
  for (int off = 128; off >= 32; off >>= 1) {
    if (tid < off) {
      float4 a = reds[tid], c = reds[tid + off];
      a.x += c.x; a.y += c.y; a.z += c.z; a.w += c.w;
      reds[tid] = a;
      float4 m0 = redm[tid], m1 = redm[tid + off];
      m0.x = fmaxf(m0.x, m1.x); m0.y = fmaxf(m0.y, m1.y);
      m0.z = fmaxf(m0.z, m1.z); m0.w = fmaxf(m0.w, m1.w);
      redm[tid] = m0;
    }
    __syncthreads();
  }

  if (tid < 32) {
    const size_t o =
        ((((size_t)b * 2 + dchunk) * SCHUNKS + sc) * 128) + (size_t)tid * 4;
    *reinterpret_cast<float4*>(psum + o) = reds[tid];
    *reinterpret_cast<float4*>(pmax + o) = redm[tid];
  }
}

// ---------------------------------------------------------------------------
// Pass 2: one block per batch. Deterministic fold of the 16 S-chunk partials,
// length = sum(mask[b,:]) reduced in LDS, emit [max(256) | mean(256)].
// ---------------------------------------------------------------------------
__global__ __launch_bounds__(256)
void pool_pass2(const float* __restrict__ mask, const float* __restrict__ psum,
                const float* __restrict__ pmax, float* __restrict__ out) {
  __shared__ float lred[256];
  const int b = blockIdx.x, tid = threadIdx.x;

  float ls = 0.f;
  for (int i = tid; i < kS; i += 256) ls += mask[(size_t)b * kS + i];
  lred[tid] = ls;
  __syncthreads();
#pragma unroll
  for (int off = 128; off >= 1; off >>= 1) {
    if (tid < off) lred[tid] += lred[tid + off];
    __syncthreads();
  }
  const float len = lred[0];

  const int dchunk = tid >> 7, dcol = tid & 127;
  const size_t base = (((size_t)b * 2 + dchunk) * SCHUNKS) * 128 + dcol;
  float s = 0.f, m = -INFINITY;
#pragma unroll
  for (int c = 0; c < SCHUNKS; ++c) {
    s += psum[base + (size_t)c * 128];
    m = fmaxf(m, pmax[base + (size_t)c * 128]);
  }
  out[(size_t)b * 512 + tid]       = m;        // max first (module order)
  out[(size_t)b * 512 + 256 + tid] = s / len;  // then mean
}

// ---------------------------------------------------------------------------
// Workspace-free fallback: one block per (b, d-chunk), full S sweep, direct
// loads, writes out[] directly. Only used if d_ws is too small.
// ---------------------------------------------------------------------------
__global__ __launch_bounds__(256)
void pool_onepass(const float* __restrict__ feats, const float* __restrict__ mask,
                  float* __restrict__ out) {
  __shared__ float4 reds[256];
  __shared__ float4 redm[256];
  __shared__ float  lred[256];

  const int tid = threadIdx.x;
  const int col = tid & 31;
  const int row = tid >> 5;
  const int dchunk = blockIdx.x & 1;
  const int b      = blockIdx.x >> 1;

  const float* fbase =
      feats + ((size_t)b * kS + row) * kD + dchunk * 128 + col * 4;

  float4 vsum = make_float4(0.f, 0.f, 0.f, 0.f);
  float4 vmax = make_float4(-INFINITY, -INFINITY, -INFINITY, -INFINITY);
  float  lsum = 0.f;

  for (int t = 0; t < kS / 8; ++t) {
    const float4 v = *reinterpret_cast<const float4*>(fbase + (size_t)t * 8 * kD);
    const float mk = mask[(size_t)b * kS + row + 8 * t];
    const bool on = mk > 0.f;
    vmax.x = fmaxf(vmax.x, on ? v.x : -INFINITY);
    vmax.y = fmaxf(vmax.y, on ? v.y : -INFINITY);
    vmax.z = fmaxf(vmax.z, on ? v.z : -INFINITY);
    vmax.w = fmaxf(vmax.w, on ? v.w : -INFINITY);
    vsum.x = fmaf(v.x, mk, vsum.x);
    vsum.y = fmaf(v.y, mk, vsum.y);
    vsum.z = fmaf(v.z, mk, vsum.z);
    vsum.w = fmaf(v.w, mk, vsum.w);
    if (col == 0) lsum += mk;   // one col per row-group accumulates length
  }

  reds[tid] = vsum;
  redm[tid] = vmax;
  lred[tid] = (col == 0) ? lsum : 0.f;
  __syncthreads();
#pragma unroll
  for (int off = 128; off >= 32; off >>= 1) {
    if (tid < off) {
      float4 a = reds[tid], c = reds[tid + off];
      a.x += c.x; a.y += c.y; a.z += c.z; a.w += c.w;
      reds[tid] = a;
      float4 m0 = redm[tid], m1 = redm[tid + off];
      m0.x = fmaxf(m0.x, m1.x); m0.y = fmaxf(m0.y, m1.y);
      m0.z = fmaxf(m0.z, m1.z); m0.w = fmaxf(m0.w, m1.w);
      redm[tid] = m0;
      lred[tid] += lred[tid + off];
    }
    __syncthreads();
  }
  const float len = lred[0];  // sum of the 8 row-group length partials

  if (tid < 32) {
    const float4 sm = reds[tid];
    const float4 mm = redm[tid];
    float* o = out + (size_t)b * 512 + dchunk * 128 + tid * 4;
    o[0] = mm.x; o[1] = mm.y; o[2] = mm.z; o[3] = mm.w;
    o[256] = sm.x / len; o[257] = sm.y / len;
    o[258] = sm.z / len; o[259] = sm.w / len;
  }
}

// ---------------------------------------------------------------------------
extern "C" void kernel_launch(void* const* d_in, const int* in_sizes, int n_in,
                              void* d_out, int out_size, void* d_ws,
                              size_t ws_size, hipStream_t stream) {
  const float* feats = (const float*)d_in[0];
  const float* mask  = (const float*)d_in[1];
  float* out = (float*)d_out;

  const size_t elems = (size_t)kB * 2 * SCHUNKS * 128;  // per partial array
  if (ws_size >= 2 * elems * sizeof(float)) {
    float* psum = (float*)d_ws;
    float* pmax = psum + elems;
    pool_pass1<<<kB * 2 * SCHUNKS, 256, 0, stream>>>(feats, mask, psum, pmax);
    pool_pass2<<<kB, 256, 0, stream>>>(mask, psum, pmax, out);
  } else {
    pool_onepass<<<kB * 2, 256, 0, stream>>>(feats, mask, out);
  }
}